// FullGNN_50105088475470
// MI455X (gfx1250) — compile-verified
//
#include <hip/hip_runtime.h>
#include <hip/hip_bf16.h>
#include <math.h>

// ---------------------------------------------------------------------------
// FullGNN for MI455X (gfx1250, wave32).
//  * Big per-edge GEMM fused as z-outer-product -> V_WMMA_F32_16X16X4_F32
//    (f32 matrix pipe; avoids materializing the 410MB per-edge weight tensor,
//    keeps every intermediate L2-resident).
//  * Lane-half-dependent A elements are loaded directly (koff folded into the
//    global load address) so no per-WMMA cndmask selects are generated.
//  * B operand staged in LDS in row-pair-interleaved layout so each WMMA B
//    fragment is one contiguous ds_load_b64 (merges to ds_load_2addr_b64).
//  * Each wave processes 2 edge tiles so every B fragment feeds 2 tiles.
//  * segment_sum via global_atomic_add_f32.
//  * Param order assumption: d_in[0..3] = node_features, edge_indices,
//    edge_features, xbatch; then params flattened jax-style (dict keys sorted).
// ---------------------------------------------------------------------------

#define NN 20000     // nodes
#define NE 100000    // edges
#define FOUT 32
#define EPS 1e-5f

typedef __attribute__((ext_vector_type(2))) float v2f;
typedef __attribute__((ext_vector_type(8))) float v8f;

// ---------------- column stats (mean / rsqrt(var+eps)), deterministic -------
__global__ __launch_bounds__(256) void colstats_kernel(
    const float* __restrict__ X, int N, int C, float* __restrict__ stats) {
  __shared__ float s1[256], s2[256];
  int c = blockIdx.x;
  float sum = 0.f, sq = 0.f;
  for (int r = threadIdx.x; r < N; r += 256) {
    float v = X[(size_t)r * C + c];
    sum += v; sq += v * v;
  }
  s1[threadIdx.x] = sum; s2[threadIdx.x] = sq;
  __syncthreads();
  for (int s = 128; s > 0; s >>= 1) {
    if (threadIdx.x < s) { s1[threadIdx.x] += s1[threadIdx.x + s]; s2[threadIdx.x] += s2[threadIdx.x + s]; }
    __syncthreads();
  }
  if (threadIdx.x == 0) {
    float m = s1[0] / (float)N;
    float var = s2[0] / (float)N - m * m;
    stats[c] = m;
    stats[C + c] = rsqrtf(var + EPS);
  }
}

// stats over virtual concat [x[src](32) | x[dst](32) | e(16)], 80 cols
__global__ __launch_bounds__(256) void edgecat_stats_kernel(
    const float* __restrict__ x, const float* __restrict__ e,
    const int* __restrict__ src, const int* __restrict__ dst,
    int E, float* __restrict__ stats) {
  __shared__ float s1[256], s2[256];
  int c = blockIdx.x;  // 0..79
  float sum = 0.f, sq = 0.f;
  for (int r = threadIdx.x; r < E; r += 256) {
    float v;
    if (c < 32)       v = x[(size_t)src[r] * 32 + c];
    else if (c < 64)  v = x[(size_t)dst[r] * 32 + (c - 32)];
    else              v = e[(size_t)r * 16 + (c - 64)];
    sum += v; sq += v * v;
  }
  s1[threadIdx.x] = sum; s2[threadIdx.x] = sq;
  __syncthreads();
  for (int s = 128; s > 0; s >>= 1) {
    if (threadIdx.x < s) { s1[threadIdx.x] += s1[threadIdx.x + s]; s2[threadIdx.x] += s2[threadIdx.x + s]; }
    __syncthreads();
  }
  if (threadIdx.x == 0) {
    float m = s1[0] / (float)E;
    float var = s2[0] / (float)E - m * m;
    stats[c] = m;
    stats[80 + c] = rsqrtf(var + EPS);
  }
}

// ---------------- elementwise BN apply --------------------------------------
__global__ __launch_bounds__(256) void bn_apply_kernel(
    const float* __restrict__ X, const float* __restrict__ stats,
    const float* __restrict__ g, const float* __restrict__ b,
    float* __restrict__ Y, int N, int C) {
  int idx = blockIdx.x * 256 + threadIdx.x;
  if (idx >= N * C) return;
  int c = idx % C;
  Y[idx] = g[c] * (X[idx] - stats[c]) * stats[C + c] + b[c];
}

__global__ __launch_bounds__(256) void zero_kernel(float* __restrict__ p, int n) {
  int i = blockIdx.x * 256 + threadIdx.x;
  if (i < n) p[i] = 0.f;
}

// ---------------- small row-wise BN->Linear(->ReLU) -------------------------
template <int D0, int D1, bool RELU>
__global__ __launch_bounds__(256) void bn_lin_kernel(
    const float* __restrict__ X, const float* __restrict__ stats,
    const float* __restrict__ g, const float* __restrict__ bb,
    const float* __restrict__ W, const float* __restrict__ bias,
    float* __restrict__ Y, int N) {
  int r = blockIdx.x * 256 + threadIdx.x;
  if (r >= N) return;
  float xn[D0];
#pragma unroll
  for (int i = 0; i < D0; ++i)
    xn[i] = g[i] * (X[(size_t)r * D0 + i] - stats[i]) * stats[D0 + i] + bb[i];
#pragma unroll
  for (int o = 0; o < D1; ++o) {
    float acc = bias[o];
#pragma unroll
    for (int i = 0; i < D0; ++i) acc = fmaf(xn[i], W[i * D1 + o], acc);
    Y[(size_t)r * D1 + o] = RELU ? (acc > 0.f ? acc : 0.f) : acc;
  }
}

// edge_upd first linear: gathered concat(x[src],x[dst],e) -> BN -> W1(80x16)+relu
__global__ __launch_bounds__(256) void upd_lin1_kernel(
    const float* __restrict__ x, const float* __restrict__ e,
    const int* __restrict__ src, const int* __restrict__ dst,
    const float* __restrict__ stats, const float* __restrict__ g,
    const float* __restrict__ bb, const float* __restrict__ W,
    const float* __restrict__ bias, float* __restrict__ Y, int E) {
  int r = blockIdx.x * 256 + threadIdx.x;
  if (r >= E) return;
  float xn[80];
  int s = src[r], d = dst[r];
#pragma unroll
  for (int i = 0; i < 32; ++i) xn[i]      = x[(size_t)s * 32 + i];
#pragma unroll
  for (int i = 0; i < 32; ++i) xn[32 + i] = x[(size_t)d * 32 + i];
#pragma unroll
  for (int i = 0; i < 16; ++i) xn[64 + i] = e[(size_t)r * 16 + i];
#pragma unroll
  for (int i = 0; i < 80; ++i) xn[i] = g[i] * (xn[i] - stats[i]) * stats[80 + i] + bb[i];
#pragma unroll
  for (int o = 0; o < 16; ++o) {
    float acc = bias[o];
#pragma unroll
    for (int i = 0; i < 80; ++i) acc = fmaf(xn[i], W[i * 16 + o], acc);
    Y[(size_t)r * 16 + o] = acc > 0.f ? acc : 0.f;
  }
}

// ---------------- fused NNConv message kernel (WMMA f32 16x16x4) ------------
// msg[e,o] = sum_{k,i} hb[e,k]*xs[e,i]*W3[k, i*32+o] + sum_i xs[e,i]*b3[i*32+o]
// One wave handles TPW=2 tiles of 16 edges. Lane l owns edge row (l&15).
// ISA 16x4 f32 A layout: lanes 0-15 carry K=Kc+0/+1, lanes 16-31 K=Kc+2/+3,
// so each lane only needs xs elements {4g+koff, 4g+koff+1}, koff=hi?2:0 —
// loaded directly from global (no per-WMMA selects).
// B operand [W3 ; b3] viewed as [NIN*NIN+NIN, 32] is staged in LDS with
// row-pair interleaving sB[(r/2)*64 + c*2 + (r&1)]: WMMA needs rows (r0,r0+1)
// at one column per register pair and r0 is always even, so each B fragment is
// one contiguous float2 -> ds_load_b64 (B0/B1 merge to ds_load_2addr_b64),
// with no repack movs. D layout: VGPR j -> rows j / j+8, col = lane&15.
// Scatter-add into agg via global f32 atomics (segment_sum).
template <int NIN, int TPW>
__global__ __launch_bounds__(256) void msg_wmma_kernel(
    const float* __restrict__ xb,     // [NN, NIN]  BN'd node features
    const float* __restrict__ h2,     // [NE, NIN]  edge_mlp hidden (pre-BN3)
    const float* __restrict__ bn3g, const float* __restrict__ bn3b,
    const float* __restrict__ hstats, // [2*NIN]
    const float* __restrict__ W3,     // [NIN, NIN*32] row-major
    const float* __restrict__ b3,     // [NIN*32]
    const int* __restrict__ src, const int* __restrict__ dst,
    float* __restrict__ agg,          // [NN, 32]
    int nEdges) {
  constexpr int KW = NIN * NIN;
  extern __shared__ float sB[];  // (KW + NIN) * 32 floats, pair-interleaved
  for (int i = threadIdx.x; i < KW * 32; i += 256) {
    int r = i >> 5, c = i & 31;
    sB[(r >> 1) * 64 + c * 2 + (r & 1)] = W3[i];
  }
  for (int i = threadIdx.x; i < NIN * 32; i += 256) {
    int r = KW + (i >> 5), c = i & 31;
    sB[(r >> 1) * 64 + c * 2 + (r & 1)] = b3[i];
  }
  __syncthreads();

  const int wave = threadIdx.x >> 5;
  const int lane = threadIdx.x & 31;
  const int tile0 = (blockIdx.x * 8 + wave) * TPW;
  if (tile0 * 16 >= nEdges) return;

  const int row = lane & 15;
  const bool hi = lane >= 16;
  const int col = row;
  const int koff = hi ? 2 : 0;
  const int phalf = hi ? 1 : 0;   // pair index offset from koff/2

  // per-tile operand rows held in registers (constant indexing after unroll)
  float hb[TPW][NIN];          // BN3(h2) row, needed in full by every lane
  float xsel[TPW][NIN / 2];    // only the half of xs this lane-half consumes
#pragma unroll
  for (int t = 0; t < TPW; ++t) {
    const int e = (tile0 + t) * 16 + row;
    const int s = src[e];
    const float* __restrict__ xr = xb + (size_t)s * NIN;
#pragma unroll
    for (int g = 0; g < NIN / 4; ++g) {
      xsel[t][2 * g + 0] = xr[4 * g + koff + 0];   // adjacent pair -> b64 load
      xsel[t][2 * g + 1] = xr[4 * g + koff + 1];
    }
#pragma unroll
    for (int i = 0; i < NIN; ++i)
      hb[t][i] = bn3g[i] * (h2[(size_t)e * NIN + i] - hstats[i]) * hstats[NIN + i] + bn3b[i];
  }

  v8f acc0[TPW], acc1[TPW];
#pragma unroll
  for (int t = 0; t < TPW; ++t) { acc0[t] = (v8f){}; acc1[t] = (v8f){}; }

#pragma unroll
  for (int k = 0; k < NIN; ++k) {
#pragma unroll
    for (int g = 0; g < NIN / 4; ++g) {
      // row pair p covers rows (2p, 2p+1); r0 = k*NIN + 4g + koff is even
      const int p = (k * NIN) / 2 + 2 * g + phalf;
      const v2f* __restrict__ bp = (const v2f*)(sB + p * 64);
      v2f B0 = bp[col];        // {row r0, row r0+1} @ col      : ds_load_b64
      v2f B1 = bp[col + 16];   // {row r0, row r0+1} @ col+16   : ds_load_b64
#pragma unroll
      for (int t = 0; t < TPW; ++t) {
        v2f A;
        A.x = hb[t][k] * xsel[t][2 * g + 0];
        A.y = hb[t][k] * xsel[t][2 * g + 1];
        acc0[t] = __builtin_amdgcn_wmma_f32_16x16x4_f32(false, A, false, B0, (short)0, acc0[t], false, false);
        acc1[t] = __builtin_amdgcn_wmma_f32_16x16x4_f32(false, A, false, B1, (short)0, acc1[t], false, false);
      }
    }
  }
  // bias rows: A = xsel directly, B rows KW..KW+NIN-1 (KW even)
#pragma unroll
  for (int g = 0; g < NIN / 4; ++g) {
    const int p = KW / 2 + 2 * g + phalf;
    const v2f* __restrict__ bp = (const v2f*)(sB + p * 64);
    v2f B0 = bp[col];
    v2f B1 = bp[col + 16];
#pragma unroll
    for (int t = 0; t < TPW; ++t) {
      v2f A;
      A.x = xsel[t][2 * g + 0];
      A.y = xsel[t][2 * g + 1];
      acc0[t] = __builtin_amdgcn_wmma_f32_16x16x4_f32(false, A, false, B0, (short)0, acc0[t], false, false);
      acc1[t] = __builtin_amdgcn_wmma_f32_16x16x4_f32(false, A, false, B1, (short)0, acc1[t], false, false);
    }
  }

  // scatter-add: lane holds D[m][col] with m = j + (hi?8:0)
#pragma unroll
  for (int t = 0; t < TPW; ++t) {
    const int mbase = (tile0 + t) * 16 + (hi ? 8 : 0);
#pragma unroll
    for (int j = 0; j < 8; ++j) {
      int d = dst[mbase + j];
      atomicAdd(&agg[(size_t)d * 32 + col], acc0[t][j]);
      atomicAdd(&agg[(size_t)d * 32 + 16 + col], acc1[t][j]);
    }
  }
}

// ---------------- node update: relu(agg + xb@root + conv_b) ----------------
template <int NIN>
__global__ __launch_bounds__(256) void node_update_kernel(
    const float* __restrict__ agg, const float* __restrict__ xb,
    const float* __restrict__ root, const float* __restrict__ convb,
    float* __restrict__ xout, int N) {
  int n = blockIdx.x * 256 + threadIdx.x;
  if (n >= N) return;
  float xr[NIN];
#pragma unroll
  for (int i = 0; i < NIN; ++i) xr[i] = xb[(size_t)n * NIN + i];
#pragma unroll
  for (int o = 0; o < 32; ++o) {
    float acc = agg[(size_t)n * 32 + o] + convb[o];
#pragma unroll
    for (int i = 0; i < NIN; ++i) acc = fmaf(xr[i], root[i * 32 + o], acc);
    xout[(size_t)n * 32 + o] = acc > 0.f ? acc : 0.f;
  }
}

// ---------------- heads -----------------------------------------------------
__global__ __launch_bounds__(256) void node_head_kernel(
    const float* __restrict__ x, const float* __restrict__ stats,
    const float* __restrict__ g, const float* __restrict__ b,
    const float* __restrict__ W, const float* __restrict__ bias,
    float* __restrict__ out, int N) {
  int n = blockIdx.x * 256 + threadIdx.x;
  if (n >= N) return;
  float xn[32];
#pragma unroll
  for (int i = 0; i < 32; ++i)
    xn[i] = g[i] * (x[(size_t)n * 32 + i] - stats[i]) * stats[32 + i] + b[i];
#pragma unroll
  for (int c = 0; c < 4; ++c) {
    float acc = bias[c];
#pragma unroll
    for (int i = 0; i < 32; ++i) acc = fmaf(xn[i], W[i * 4 + c], acc);
    out[(size_t)n * 4 + c] = (c < 3) ? tanhf(acc) : 2.f / (1.f + expf(-acc));
  }
}

__global__ __launch_bounds__(256) void edge_head_kernel(
    const float* __restrict__ e, const float* __restrict__ W,
    const float* __restrict__ bias, float* __restrict__ out, int E) {
  int r = blockIdx.x * 256 + threadIdx.x;
  if (r >= E) return;
#pragma unroll
  for (int c = 0; c < 2; ++c) {
    float acc = bias[c];
#pragma unroll
    for (int i = 0; i < 16; ++i) acc = fmaf(e[(size_t)r * 16 + i], W[i * 2 + c], acc);
    out[(size_t)r * 2 + c] = acc;
  }
}

// ---------------------------------------------------------------------------
extern "C" void kernel_launch(void* const* d_in, const int* in_sizes, int n_in,
                              void* d_out, int out_size, void* d_ws, size_t ws_size,
                              hipStream_t stream) {
  (void)in_sizes; (void)n_in; (void)out_size; (void)ws_size;
  const float* x0 = (const float*)d_in[0];
  const int* eidx = (const int*)d_in[1];
  const float* e0 = (const float*)d_in[2];
  const int* src = eidx;
  const int* dst = eidx + NE;

  // params flattened (jax tree order, dict keys sorted):
  // 4:ep_W 5:ep_b 6..89:layers(28 leaves each) 90:np_W 91:np_b 92:np_bn_b 93:np_bn_g
  auto P = [&](int i) { return (const float*)d_in[i]; };

  float* ws = (float*)d_ws;
  float* xb   = ws;               // 640000
  float* xA   = ws + 640000;      // 640000
  float* xB   = ws + 1280000;     // 640000
  float* agg  = ws + 1920000;     // 640000
  float* eP   = ws + 2560000;     // 1600000
  float* eQ   = ws + 4160000;     // 1600000
  float* h1   = ws + 5760000;     // 3200000 (reused as he1)
  float* h2   = ws + 8960000;     // 3200000 (reused as he2)
  float* st   = ws + 12160000;    // 1024
  float* nodeS = st, *eS1 = st + 64, *eS2 = st + 128, *eS3 = st + 192;
  float* catS = st + 256, *uS2 = st + 448, *uS3 = st + 512, *finS = st + 576;

  const float* xcur = x0;
  const float* ecur = e0;
  float* xnext = xA;
  float* enext = eP;

  const int EB = (NE + 255) / 256;
  const int NB = (NN + 255) / 256;

  for (int l = 0; l < 3; ++l) {
    const int NIN = l ? 32 : 16;
    const int L = 6 + 28 * l;
    const float *bn_node_b = P(L+0), *bn_node_g = P(L+1), *conv_b = P(L+2);
    const float *W1 = P(L+3), *W2 = P(L+4), *W3 = P(L+5);
    const float *b1 = P(L+6), *b2 = P(L+7), *b3 = P(L+8);
    const float *bn1b = P(L+9), *bn1g = P(L+10), *bn2b = P(L+11), *bn2g = P(L+12);
    const float *bn3b = P(L+13), *bn3g = P(L+14);
    const float *uW1 = P(L+15), *uW2 = P(L+16), *uW3 = P(L+17);
    const float *ub1 = P(L+18), *ub2 = P(L+19), *ub3 = P(L+20);
    const float *ubn1b = P(L+21), *ubn1g = P(L+22), *ubn2b = P(L+23), *ubn2g = P(L+24);
    const float *ubn3b = P(L+25), *ubn3g = P(L+26);
    const float *root = P(L+27);

    // node BN -> xb
    colstats_kernel<<<NIN, 256, 0, stream>>>(xcur, NN, NIN, nodeS);
    bn_apply_kernel<<<(NN * NIN + 255) / 256, 256, 0, stream>>>(
        xcur, nodeS, bn_node_g, bn_node_b, xb, NN, NIN);

    // edge MLP hidden layers
    colstats_kernel<<<16, 256, 0, stream>>>(ecur, NE, 16, eS1);
    if (NIN == 16)
      bn_lin_kernel<16, 16, true><<<EB, 256, 0, stream>>>(ecur, eS1, bn1g, bn1b, W1, b1, h1, NE);
    else
      bn_lin_kernel<16, 32, true><<<EB, 256, 0, stream>>>(ecur, eS1, bn1g, bn1b, W1, b1, h1, NE);
    colstats_kernel<<<NIN, 256, 0, stream>>>(h1, NE, NIN, eS2);
    if (NIN == 16)
      bn_lin_kernel<16, 16, true><<<EB, 256, 0, stream>>>(h1, eS2, bn2g, bn2b, W2, b2, h2, NE);
    else
      bn_lin_kernel<32, 32, true><<<EB, 256, 0, stream>>>(h1, eS2, bn2g, bn2b, W2, b2, h2, NE);
    colstats_kernel<<<NIN, 256, 0, stream>>>(h2, NE, NIN, eS3);

    // fused NNConv message + scatter-add
    zero_kernel<<<(NN * 32 + 255) / 256, 256, 0, stream>>>(agg, NN * 32);
    const int tiles = NE / 16;                    // 6250, exact
    const int grid = (tiles / 2 + 7) / 8;         // 2 tiles/wave, 8 waves/block
    if (NIN == 16) {
      size_t smem = (size_t)(16 * 16 + 16) * 32 * sizeof(float);
      msg_wmma_kernel<16, 2><<<grid, 256, smem, stream>>>(
          xb, h2, bn3g, bn3b, eS3, W3, b3, src, dst, agg, NE);
    } else {
      size_t smem = (size_t)(32 * 32 + 32) * 32 * sizeof(float);
      msg_wmma_kernel<32, 2><<<grid, 256, smem, stream>>>(
          xb, h2, bn3g, bn3b, eS3, W3, b3, src, dst, agg, NE);
    }

    // x = relu(agg + xb@root + conv_b)
    if (NIN == 16)
      node_update_kernel<16><<<NB, 256, 0, stream>>>(agg, xb, root, conv_b, xnext, NN);
    else
      node_update_kernel<32><<<NB, 256, 0, stream>>>(agg, xb, root, conv_b, xnext, NN);

    // edge update MLP on concat(x[src], x[dst], e)
    edgecat_stats_kernel<<<80, 256, 0, stream>>>(xnext, ecur, src, dst, NE, catS);
    upd_lin1_kernel<<<EB, 256, 0, stream>>>(xnext, ecur, src, dst, catS, ubn1g, ubn1b,
                                            uW1, ub1, h1, NE);
    colstats_kernel<<<16, 256, 0, stream>>>(h1, NE, 16, uS2);
    bn_lin_kernel<16, 16, true><<<EB, 256, 0, stream>>>(h1, uS2, ubn2g, ubn2b, uW2, ub2, h2, NE);
    colstats_kernel<<<16, 256, 0, stream>>>(h2, NE, 16, uS3);
    bn_lin_kernel<16, 16, false><<<EB, 256, 0, stream>>>(h2, uS3, ubn3g, ubn3b, uW3, ub3, enext, NE);

    xcur = xnext; xnext = (xnext == xA) ? xB : xA;
    ecur = enext; enext = (enext == eP) ? eQ : eP;
  }

  // output heads
  colstats_kernel<<<32, 256, 0, stream>>>(xcur, NN, 32, finS);
  node_head_kernel<<<NB, 256, 0, stream>>>(xcur, finS, P(93) /*np_bn_g*/, P(92) /*np_bn_b*/,
                                           P(90) /*np_W*/, P(91) /*np_b*/, (float*)d_out, NN);
  edge_head_kernel<<<EB, 256, 0, stream>>>(ecur, P(4) /*ep_W*/, P(5) /*ep_b*/,
                                           (float*)d_out + 4 * NN, NE);
}